// CCELossFast_66649302499841
// MI455X (gfx1250) — compile-verified
//
#include <hip/hip_runtime.h>
#include <math.h>

#define NBINS 10
#define WAVE 32
#define WAVES_PER_BLOCK 8
#define BLOCK (WAVE * WAVES_PER_BLOCK)
#define MAX_SLOTS 8            // supports C <= 4*32*8 = 1024 classes
#define ROWBUF_V4 250          // float4 slots per staged row (C=1000 -> 250)
#define MAX_C 1024
#define LOG2E 1.4426950408889634f
#define EPS 1e-13f

// ---- CDNA5 async global->LDS helpers (gfx1250) --------------------------

__device__ __forceinline__ unsigned lds_addr32(const void* p) {
    // generic -> LDS(3) addrspace cast, then take the 32-bit LDS offset
    return (unsigned)(unsigned long long)(__attribute__((address_space(3))) const char*)p;
}

__device__ __forceinline__ void async_copy_b128(unsigned lds_off, const void* gaddr) {
    asm volatile("global_load_async_to_lds_b128 %0, %1, off"
                 :: "v"(lds_off), "v"((unsigned long long)gaddr)
                 : "memory");
}
__device__ __forceinline__ void wait_async_le8() { asm volatile("s_wait_asynccnt 8" ::: "memory"); }
__device__ __forceinline__ void wait_async_0()  { asm volatile("s_wait_asynccnt 0" ::: "memory"); }

// ---- Kernel 1: softmax + binning + (class,bin) histogram ----------------
//
// One wave per row, rows double-buffered in LDS via async DMA (ASYNCcnt).
// Bin 0 is privatized: per-lane conf accumulators in registers, combined
// per-block in LDS, flushed once. no_pred[c][0] = B + rare-event corrections.
// CNV4: compile-time float4-count per row (250 for C=1000); 0 = dynamic.

template <int CNV4>
__global__ void __launch_bounds__(BLOCK)
k_hist(const float* __restrict__ x, const int* __restrict__ tgt,
       float* __restrict__ np, float* __restrict__ na, float* __restrict__ cf,
       int B, int C) {
    __shared__ float4 stagebuf[WAVES_PER_BLOCK][2][ROWBUF_V4];
    __shared__ float  conf0[MAX_C];

    const int lane   = threadIdx.x & (WAVE - 1);
    const int wloc   = threadIdx.x >> 5;
    const int wid    = blockIdx.x * WAVES_PER_BLOCK + wloc;
    const int nwaves = gridDim.x * WAVES_PER_BLOCK;
    const int nv4    = CNV4 ? CNV4 : (C >> 2);

    for (int i = threadIdx.x; i < C; i += BLOCK) conf0[i] = 0.0f;
    __syncthreads();

    // compile-time slot validity for the specialized instantiation
#define SLOT_VALID(k) (CNV4 ? (((k) + 1) * WAVE <= CNV4 || lane < (CNV4 - (k) * WAVE)) \
                            : ((lane + WAVE * (k)) < nv4))

    float acc[MAX_SLOTS][4];               // private bin-0 conf sums
#pragma unroll
    for (int k = 0; k < MAX_SLOTS; ++k)
#pragma unroll
        for (int j = 0; j < 4; ++j) acc[k][j] = 0.0f;

    const int r0 = wid;
    if (r0 < B) {
        const float4* grow = (const float4*)(x + (size_t)r0 * C);
        float4* lbuf = &stagebuf[wloc][0][0];
#pragma unroll
        for (int k = 0; k < MAX_SLOTS; ++k) {
            int v = lane + WAVE * k;
            if (SLOT_VALID(k)) async_copy_b128(lds_addr32(lbuf + v), (const void*)(grow + v));
        }
    }

    int parity = 0;
    for (int r = r0; r < B; r += nwaves) {
        const int  rn   = r + nwaves;
        const bool more = rn < B;
        if (more) {
            const float4* grow = (const float4*)(x + (size_t)rn * C);
            float4* lbuf = &stagebuf[wloc][parity ^ 1][0];
#pragma unroll
            for (int k = 0; k < MAX_SLOTS; ++k) {
                int v = lane + WAVE * k;
                if (SLOT_VALID(k)) async_copy_b128(lds_addr32(lbuf + v), (const void*)(grow + v));
            }
        }
        if (more) wait_async_le8(); else wait_async_0();   // current row staged

        const float4* buf = &stagebuf[wloc][parity][0];

        // pass 1: row from LDS, lane-local + wave max
        float v[MAX_SLOTS][4];
        float m = -__builtin_inff();
#pragma unroll
        for (int k = 0; k < MAX_SLOTS; ++k) {
            if (SLOT_VALID(k)) {
                float4 t = buf[lane + WAVE * k];
                v[k][0] = t.x; v[k][1] = t.y; v[k][2] = t.z; v[k][3] = t.w;
                m = fmaxf(m, fmaxf(fmaxf(t.x, t.y), fmaxf(t.z, t.w)));
            }
        }
#pragma unroll
        for (int off = 16; off >= 1; off >>= 1) m = fmaxf(m, __shfl_xor(m, off, WAVE));

        // pass 2: exp (v_exp_f32) + wave sum
        float s = 0.0f;
#pragma unroll
        for (int k = 0; k < MAX_SLOTS; ++k) {
            if (SLOT_VALID(k)) {
#pragma unroll
                for (int j = 0; j < 4; ++j) {
                    float e = __builtin_amdgcn_exp2f((v[k][j] - m) * LOG2E);
                    v[k][j] = e;
                    s += e;
                }
            }
        }
#pragma unroll
        for (int off = 16; off >= 1; off >>= 1) s += __shfl_xor(s, off, WAVE);
        const float inv = 1.0f / s;
        const int   ct  = tgt[r];

        // pass 3: probs, bins; rare events behind one cold branch
#pragma unroll
        for (int k = 0; k < MAX_SLOTS; ++k) {
            if (SLOT_VALID(k)) {
#pragma unroll
                for (int j = 0; j < 4; ++j) {
                    const float p = v[k][j] * inv;
                    const int   c = 4 * (lane + WAVE * k) + j;
                    acc[k][j] += p;
                    int bin = (int)ceilf(p * 10.0f) - 1;
                    bin = bin < 0 ? 0 : (bin > NBINS - 1 ? NBINS - 1 : bin);
                    const bool rare = (bin != 0);
                    const bool hit  = (c == ct);
                    if (__builtin_expect(rare || hit, 0)) {
                        if (rare) {                        // p > 0.1 (uncommon)
                            acc[k][j] -= p;
                            atomicAdd(&cf[c * NBINS + bin],  p);
                            atomicAdd(&np[c * NBINS + bin],  1.0f);
                            atomicAdd(&np[c * NBINS + 0],   -1.0f);
                        }
                        if (hit)                            // one lane per row
                            atomicAdd(&na[c * NBINS + bin], 1.0f);
                    }
                }
            }
        }
        parity ^= 1;
    }

    // combine per-wave bin-0 conf accumulators in block LDS (ds_add_f32)
#pragma unroll
    for (int k = 0; k < MAX_SLOTS; ++k) {
        if (SLOT_VALID(k)) {
#pragma unroll
            for (int j = 0; j < 4; ++j) {
                if (acc[k][j] != 0.0f)
                    atomicAdd(&conf0[4 * (lane + WAVE * k) + j], acc[k][j]);
            }
        }
    }
    __syncthreads();

    // one cooperative global flush per block
    for (int i = threadIdx.x; i < C; i += BLOCK) {
        float a = conf0[i];
        if (a != 0.0f) atomicAdd(&cf[i * NBINS], a);
    }
#undef SLOT_VALID
}

// ---- Kernel 2: 10000-segment finalize -> scalar loss --------------------

__global__ void k_final(const float* __restrict__ np, const float* __restrict__ na,
                        const float* __restrict__ cf, float* __restrict__ out,
                        int B, int C) {
    __shared__ float red[256];
    const int   n_seg = C * NBINS;
    const float total = (float)B * (float)C;   // exact for 262144*1000
    float local = 0.0f;
    for (int s = threadIdx.x; s < n_seg; s += blockDim.x) {
        float n = np[s] + (((s % NBINS) == 0) ? (float)B : 0.0f);  // bin-0 analytic base
        float d = fabsf(na[s] - cf[s]);
        local += d * n / ((n + EPS) * total);
    }
    red[threadIdx.x] = local;
    __syncthreads();
    for (int st = 128; st > 0; st >>= 1) {
        if ((int)threadIdx.x < st) red[threadIdx.x] += red[threadIdx.x + st];
        __syncthreads();
    }
    if (threadIdx.x == 0) out[0] = red[0];
}

// ---- Host launch --------------------------------------------------------

extern "C" void kernel_launch(void* const* d_in, const int* in_sizes, int n_in,
                              void* d_out, int out_size, void* d_ws, size_t ws_size,
                              hipStream_t stream) {
    const float* x   = (const float*)d_in[0];
    const int*   tgt = (const int*)d_in[1];
    const int B = in_sizes[1];
    const int C = in_sizes[0] / B;      // 1000

    float* np = (float*)d_ws;
    float* na = np + (size_t)C * NBINS;
    float* cf = na + (size_t)C * NBINS;

    hipMemsetAsync(d_ws, 0, (size_t)3 * C * NBINS * sizeof(float), stream);

    const int blocks = 512;             // 4096 waves -> 64 rows/wave
    if (C == 1000) {
        k_hist<250><<<blocks, BLOCK, 0, stream>>>(x, tgt, np, na, cf, B, C);
    } else {
        k_hist<0><<<blocks, BLOCK, 0, stream>>>(x, tgt, np, na, cf, B, C);
    }
    k_final<<<1, 256, 0, stream>>>(np, na, cf, (float*)d_out, B, C);
}